// DSPLTnet_40827959116196
// MI455X (gfx1250) — compile-verified
//
#include <hip/hip_runtime.h>
#include <math.h>

typedef __attribute__((ext_vector_type(2))) float v2f;
typedef __attribute__((ext_vector_type(8))) float v8f;

#define PROTO_M_F 0.99f
#define NUM_CLASSES 1000
#define OUT_DIM 512
#define BATCH 16384
#define APAD 528           // 513 padded up to multiple of 16 (and of 4 for K tiles)
#define K_TAYLOR 18        // ||A||_1 ~ 0.2 -> fp32-exact well before 18 terms

// ---------------------------------------------------------------------------
// K1: A = (wp - wn)^2, zero-padded to 528x528 row-major.
// ---------------------------------------------------------------------------
__global__ void build_A_kernel(const float* __restrict__ wp,
                               const float* __restrict__ wn,
                               float* __restrict__ A) {
  int idx = blockIdx.x * blockDim.x + threadIdx.x;
  if (idx >= APAD * APAD) return;
  int r = idx / APAD, c = idx % APAD;
  float v = 0.0f;
  if (r < 513 && c < 513) {
    float d = wp[r * 513 + c] - wn[r * 513 + c];
    v = d * d;
  }
  A[idx] = v;
}

// ---------------------------------------------------------------------------
// K2: mask = (expm(A) e_512)[:512] via Horner Taylor: r = e + (A r)/k.
// Matvec done as 528x528 * 528x16 panel (only col 0 live) with
// v_wmma_f32_16x16x4_f32. Single workgroup, 32 waves, 33 row strips,
// x-column double-buffered in LDS.
// ---------------------------------------------------------------------------
__global__ __launch_bounds__(1024) void expm_col_kernel(const float* __restrict__ A,
                                                        float* __restrict__ mask_out) {
  __shared__ float xcol[2][APAD + 16];
  const int tid  = threadIdx.x;
  const int wave = tid >> 5;
  const int lane = tid & 31;
  const int half = lane >> 4;   // 0: lanes 0-15, 1: lanes 16-31
  const int r    = lane & 15;

  for (int i = tid; i < APAD + 16; i += 1024) {
    xcol[0][i] = (i == 512) ? 1.0f : 0.0f;   // e_512
    xcol[1][i] = 0.0f;
  }
  __syncthreads();

  int cur = 0;
  for (int k = K_TAYLOR; k >= 1; --k) {
    const float invk = 1.0f / (float)k;
    for (int s = wave; s < APAD / 16; s += 32) {   // wave-uniform strip loop
      v8f acc = {};
      const int row = s * 16 + r;
      const float* Arow = A + row * APAD;
      for (int kt = 0; kt < APAD / 4; ++kt) {
        const int k0 = kt * 4 + 2 * half;
        v2f a;                              // A 16x4: VGPR j <-> K = 2*half + j
        a.x = Arow[k0];
        a.y = Arow[k0 + 1];
        float b0 = xcol[cur][k0];           // B 4x16: only column N==0 nonzero
        float b1 = xcol[cur][k0 + 1];
        v2f b;
        b.x = (r == 0) ? b0 : 0.0f;
        b.y = (r == 0) ? b1 : 0.0f;
        acc = __builtin_amdgcn_wmma_f32_16x16x4_f32(
            /*neg_a=*/false, a, /*neg_b=*/false, b,
            /*c_mod=*/(short)0, acc, /*reuse_a=*/false, /*reuse_b=*/false);
      }
      // Column N==0 of C/D lives in lanes 0 (M=0..7) and 16 (M=8..15).
      if (r == 0) {
#pragma unroll
        for (int i2 = 0; i2 < 8; ++i2) {
          int ro = s * 16 + half * 8 + i2;
          float e = (ro == 512) ? 1.0f : 0.0f;
          xcol[cur ^ 1][ro] = e + acc[i2] * invk;
        }
      }
    }
    __syncthreads();
    cur ^= 1;
  }
  for (int i = tid; i < OUT_DIM; i += 1024) mask_out[i] = xcol[cur][i];
}

// ---------------------------------------------------------------------------
// K3a: per-chunk class histograms (16 chunks of 1024 samples).
// ---------------------------------------------------------------------------
__global__ __launch_bounds__(256) void hist_kernel(const int* __restrict__ y,
                                                   int* __restrict__ chunk_hist) {
  __shared__ int lh[NUM_CLASSES];
  for (int i = threadIdx.x; i < NUM_CLASSES; i += 256) lh[i] = 0;
  __syncthreads();
  const int base = blockIdx.x * 1024;
  for (int i = threadIdx.x; i < 1024; i += 256) atomicAdd(&lh[y[base + i]], 1);
  __syncthreads();
  for (int i = threadIdx.x; i < NUM_CLASSES; i += 256)
    chunk_hist[blockIdx.x * NUM_CLASSES + i] = lh[i];
}

__global__ void zero_counts_kernel(int* __restrict__ counts) {
  int i = blockIdx.x * blockDim.x + threadIdx.x;
  if (i < NUM_CLASSES) counts[i] = 0;
}

// ---------------------------------------------------------------------------
// K3b: rank_i = #{j<i : y_j==y_i}; w_i = (1-m) * m^(count-1-rank);
//      first occurrence records count[class].
// ---------------------------------------------------------------------------
__global__ __launch_bounds__(1024) void rank_kernel(const int* __restrict__ y,
                                                    const int* __restrict__ chunk_hist,
                                                    float* __restrict__ w,
                                                    int* __restrict__ counts) {
  __shared__ int ly[1024];
  const int chunk = blockIdx.x;
  const int tid = threadIdx.x;
  ly[tid] = y[chunk * 1024 + tid];
  __syncthreads();
  const int yi = ly[tid];
  int pre = 0, total = 0;
  for (int b = 0; b < 16; ++b) {
    int h = chunk_hist[b * NUM_CLASSES + yi];
    total += h;
    if (b < chunk) pre += h;
  }
  int rin = 0;
  for (int j = 0; j < tid; ++j) rin += (ly[j] == yi) ? 1 : 0;
  const int rank = pre + rin;
  const int suffix = total - 1 - rank;
  w[chunk * 1024 + tid] = (1.0f - PROTO_M_F) * powf(PROTO_M_F, (float)suffix);
  if (rank == 0) counts[yi] = total;
}

// ---------------------------------------------------------------------------
// K4a: out_p = m^count * proto ; out_py = m^count * proto_y  (overwrites d_out)
// ---------------------------------------------------------------------------
__global__ void init_out_kernel(const float* __restrict__ proto,
                                const float* __restrict__ proto_y,
                                const int* __restrict__ counts,
                                float* __restrict__ out) {
  int idx = blockIdx.x * blockDim.x + threadIdx.x;
  if (idx >= NUM_CLASSES * OUT_DIM) return;
  int c = idx >> 9;
  float scale = powf(PROTO_M_F, (float)counts[c]);
  out[idx] = scale * proto[idx];
  out[NUM_CLASSES * OUT_DIM + idx] = scale * proto_y[idx];
}

// ---------------------------------------------------------------------------
// K4b: per-sample: row-normalize f and mask*f_aug, atomically scatter
//      w_i * normalized rows into out[y_i]. One block per sample, float4.
// ---------------------------------------------------------------------------
__global__ __launch_bounds__(128) void scatter_kernel(const float* __restrict__ f,
                                                      const float* __restrict__ fa,
                                                      const int* __restrict__ y,
                                                      const float* __restrict__ mask,
                                                      const float* __restrict__ w,
                                                      float* __restrict__ out) {
  const int i = blockIdx.x;
  const int tid = threadIdx.x;
  const int lane = tid & 31;
  const int wv = tid >> 5;

  float4 fv = ((const float4*)(f + (size_t)i * OUT_DIM))[tid];
  float4 mv = ((const float4*)mask)[tid];
  float4 av = ((const float4*)(fa + (size_t)i * OUT_DIM))[tid];
  av.x *= mv.x; av.y *= mv.y; av.z *= mv.z; av.w *= mv.w;

  float s1 = fv.x * fv.x + fv.y * fv.y + fv.z * fv.z + fv.w * fv.w;
  float s2 = av.x * av.x + av.y * av.y + av.z * av.z + av.w * av.w;
#pragma unroll
  for (int off = 16; off > 0; off >>= 1) {
    s1 += __shfl_xor(s1, off, 32);
    s2 += __shfl_xor(s2, off, 32);
  }
  __shared__ float red[2][4];
  if (lane == 0) { red[0][wv] = s1; red[1][wv] = s2; }
  __syncthreads();
  s1 = red[0][0] + red[0][1] + red[0][2] + red[0][3];
  s2 = red[1][0] + red[1][1] + red[1][2] + red[1][3];

  const float inv1 = 1.0f / fmaxf(sqrtf(s1), 1e-12f);
  const float inv2 = 1.0f / fmaxf(sqrtf(s2), 1e-12f);
  const float wi = w[i];
  const int c = y[i];
  float* po  = out + (size_t)c * OUT_DIM + tid * 4;
  float* pyo = out + (size_t)NUM_CLASSES * OUT_DIM + (size_t)c * OUT_DIM + tid * 4;
  const float a1 = wi * inv1, a2 = wi * inv2;
  unsafeAtomicAdd(&po[0], a1 * fv.x);
  unsafeAtomicAdd(&po[1], a1 * fv.y);
  unsafeAtomicAdd(&po[2], a1 * fv.z);
  unsafeAtomicAdd(&po[3], a1 * fv.w);
  unsafeAtomicAdd(&pyo[0], a2 * av.x);
  unsafeAtomicAdd(&pyo[1], a2 * av.y);
  unsafeAtomicAdd(&pyo[2], a2 * av.z);
  unsafeAtomicAdd(&pyo[3], a2 * av.w);
}

// ---------------------------------------------------------------------------
// K5: final per-row L2 normalization of the 2000 output rows (in place).
// ---------------------------------------------------------------------------
__global__ __launch_bounds__(128) void norm_kernel(float* __restrict__ out) {
  const int row = blockIdx.x;             // 0..1999
  const int tid = threadIdx.x;
  const int lane = tid & 31;
  const int wv = tid >> 5;
  float4* p = (float4*)(out + (size_t)row * OUT_DIM);
  float4 v = p[tid];
  float s = v.x * v.x + v.y * v.y + v.z * v.z + v.w * v.w;
#pragma unroll
  for (int off = 16; off > 0; off >>= 1) s += __shfl_xor(s, off, 32);
  __shared__ float red[4];
  if (lane == 0) red[wv] = s;
  __syncthreads();
  s = red[0] + red[1] + red[2] + red[3];
  const float inv = 1.0f / fmaxf(sqrtf(s), 1e-12f);
  v.x *= inv; v.y *= inv; v.z *= inv; v.w *= inv;
  p[tid] = v;
}

// ---------------------------------------------------------------------------
extern "C" void kernel_launch(void* const* d_in, const int* in_sizes, int n_in,
                              void* d_out, int out_size, void* d_ws, size_t ws_size,
                              hipStream_t stream) {
  const float* f       = (const float*)d_in[0];
  const float* f_aug   = (const float*)d_in[1];
  const int*   y       = (const int*)d_in[2];
  const float* proto   = (const float*)d_in[3];
  const float* proto_y = (const float*)d_in[4];
  const float* wpos    = (const float*)d_in[5];
  const float* wneg    = (const float*)d_in[6];
  float* out = (float*)d_out;

  char* ws = (char*)d_ws;
  float* A      = (float*)(ws);                       // 528*528*4 = 1,115,136 B
  float* mask   = (float*)(ws + 1115136);             // 512*4
  float* w      = (float*)(ws + 1117184);             // 16384*4
  int*   counts = (int*)  (ws + 1182720);             // 1000*4
  int*   chist  = (int*)  (ws + 1186720);             // 16*1000*4

  build_A_kernel<<<(APAD * APAD + 255) / 256, 256, 0, stream>>>(wpos, wneg, A);
  zero_counts_kernel<<<4, 256, 0, stream>>>(counts);
  hist_kernel<<<16, 256, 0, stream>>>(y, chist);
  rank_kernel<<<16, 1024, 0, stream>>>(y, chist, w, counts);
  expm_col_kernel<<<1, 1024, 0, stream>>>(A, mask);
  init_out_kernel<<<(NUM_CLASSES * OUT_DIM + 255) / 256, 256, 0, stream>>>(proto, proto_y, counts, out);
  scatter_kernel<<<BATCH, 128, 0, stream>>>(f, f_aug, y, mask, w, out);
  norm_kernel<<<2 * NUM_CLASSES, 128, 0, stream>>>(out);
}